// sparseAttention_88484916232620
// MI455X (gfx1250) — compile-verified
//
#include <hip/hip_runtime.h>
#include <math.h>

typedef float v2f __attribute__((ext_vector_type(2)));
typedef float v8f __attribute__((ext_vector_type(8)));

#define HEADS   8
#define DK      32
#define RATE    2
#define SEQ     2046
#define NPAD    2048
#define MROWS   1024      // rows per (b,h,r) group
#define DMODEL  256
#define BATCH   4

#define YSTRIDE 34        // LDS row stride for Y (even for 8B align, bank-spread)
#define PSTRIDE 18        // LDS row stride for P scratch

// ---------------------------------------------------------------------------
// Kernel 1: qkv[8192 x 256] = xp @ W + b, xp rows >= SEQ (per batch) are zero.
// One wave computes one 16x16 output tile with V_WMMA_F32_16X16X4_F32.
// Branch-free: padded rows load from a safe address and are masked to zero,
// so the K-loop is a pure load-clause + WMMA stream (EXEC stays all-ones).
// ---------------------------------------------------------------------------
__global__ __launch_bounds__(32)
void qkv_gemm_kernel(const float* __restrict__ x,
                     const float* __restrict__ W,
                     const float* __restrict__ bias,
                     float* __restrict__ qkv)
{
    const int lane = threadIdx.x;
    const int half = lane >> 4;      // 0/1
    const int l16  = lane & 15;

    const int mtile = blockIdx.x;    // 0..511  (8192/16)
    const int ntile = blockIdx.y;    // 0..15   (256/16)

    const int row = mtile * 16 + l16;        // global qkv row, 0..8191
    const int bb  = row >> 11;               // /2048
    const int n   = row & (NPAD - 1);
    const bool valid = (n < SEQ);
    const int col = ntile * 16 + l16;

    // padded rows: read row 0 of the batch (always in-bounds), mask to zero
    const float mask = valid ? 1.0f : 0.0f;
    const float* xrow = x + ((size_t)(bb * SEQ + (valid ? n : 0))) * DMODEL;

    // C init = bias (depends only on column = lane)
    const float bv = bias[col];
    v8f c;
    #pragma unroll
    for (int v = 0; v < 8; ++v) c[v] = bv;

    #pragma unroll 4
    for (int k0 = 0; k0 < DMODEL; k0 += 4) {
        const int k = k0 + 2 * half;
        const float2 av = *(const float2*)(xrow + k);
        v2f a;
        a[0] = av.x * mask;
        a[1] = av.y * mask;
        v2f bm;
        bm[0] = W[(size_t)k * DMODEL + col];
        bm[1] = W[(size_t)(k + 1) * DMODEL + col];
        c = __builtin_amdgcn_wmma_f32_16x16x4_f32(false, a, false, bm,
                                                  (short)0, c, false, false);
    }

    // C/D layout: vgpr v, lanes 0-15 -> M=v, lanes 16-31 -> M=v+8, N=l16
    #pragma unroll
    for (int v = 0; v < 8; ++v) {
        const int rrow = mtile * 16 + v + 8 * half;
        qkv[(size_t)rrow * DMODEL + col] = c[v];
    }
}

// ---------------------------------------------------------------------------
// Kernel 2: per-group attention.
// grid = (8 chunks, 64 groups), block = 256 threads (8 waves).
// Each wave owns one 16-row query block; Y (1024x32) staged in LDS.
// ---------------------------------------------------------------------------
__global__ __launch_bounds__(256)
void attn_kernel(const float* __restrict__ qkv, float* __restrict__ out)
{
    extern __shared__ float lds[];
    float* Ys = lds;                              // MROWS * YSTRIDE floats
    float* Ps = lds + MROWS * YSTRIDE;            // 8 * 16 * PSTRIDE floats

    const int tid  = threadIdx.x;
    const int wave = tid >> 5;
    const int lane = tid & 31;
    const int half = lane >> 4;
    const int l16  = lane & 15;

    const int g  = blockIdx.y;       // 0..63
    const int bb = g >> 4;           // batch
    const int h  = (g >> 1) & 7;     // head
    const int r  = g & 1;            // phase

    // ---- stage Y: row i of group = qkv[bb, 2*i + r, h*32 : h*32+32]
    {
        const int dcol  = (tid & 15) * 2;
        const int rbase = tid >> 4;              // 0..15
        for (int i = rbase; i < MROWS; i += 16) {
            const float* src = qkv + ((size_t)(bb * NPAD + 2 * i + r)) * DMODEL
                                   + h * DK + dcol;
            const float2 v = *(const float2*)src;
            *(float2*)&Ys[i * YSTRIDE + dcol] = v;
        }
    }
    __syncthreads();

    float* Pw = Ps + wave * 16 * PSTRIDE;

    const int qb    = blockIdx.x * 8 + wave;     // query block 0..63
    const int qbase = qb * 16;

    const float scale = 0.17677669529663687f;    // 1/sqrt(32)

    float mrow[8], lrow[8];
    v8f o0, o1;
    #pragma unroll
    for (int v = 0; v < 8; ++v) {
        mrow[v] = -3.0e38f; lrow[v] = 0.0f;
        o0[v] = 0.0f; o1[v] = 0.0f;
    }

    // ---- main loop over 64 key blocks: online softmax
    for (int kb = 0; kb < 64; ++kb) {
        const int kbase = kb * 16;

        // S tile = Q(16x32) * K(16x32)^T via 8 x WMMA f32 16x16x4
        v8f s;
        #pragma unroll
        for (int v = 0; v < 8; ++v) s[v] = 0.0f;
        #pragma unroll
        for (int d0 = 0; d0 < DK; d0 += 4) {
            const int d = d0 + 2 * half;
            v2f a, b;
            const float2 av = *(const float2*)&Ys[(qbase + l16) * YSTRIDE + d];
            const float2 bv = *(const float2*)&Ys[(kbase + l16) * YSTRIDE + d];
            a[0] = av.x; a[1] = av.y;
            b[0] = bv.x; b[1] = bv.y;
            s = __builtin_amdgcn_wmma_f32_16x16x4_f32(false, a, false, b,
                                                      (short)0, s, false, false);
        }

        // per-row (vgpr v -> row v + 8*half) online softmax update
        #pragma unroll
        for (int v = 0; v < 8; ++v) {
            const float sv = s[v] * scale;
            float rmax = sv;
            #pragma unroll
            for (int mk = 1; mk < 16; mk <<= 1)
                rmax = fmaxf(rmax, __shfl_xor(rmax, mk, 32));
            const float newm = fmaxf(mrow[v], rmax);
            const float corr = __expf(mrow[v] - newm);
            const float p    = __expf(sv - newm);
            float rsum = p;
            #pragma unroll
            for (int mk = 1; mk < 16; mk <<= 1)
                rsum += __shfl_xor(rsum, mk, 32);
            lrow[v] = lrow[v] * corr + rsum;
            mrow[v] = newm;
            o0[v] *= corr;
            o1[v] *= corr;
            // stash P in D-layout: row = v + 8*half, col = l16
            Pw[(v + 8 * half) * PSTRIDE + l16] = p;
        }

        // cross-lane exchange through LDS: make P stores visible to A-layout loads
        asm volatile("s_wait_dscnt 0x0" ::: "memory");

        // O += P(16x16) * V(16x32), V rows = key rows from LDS
        #pragma unroll
        for (int k0 = 0; k0 < 16; k0 += 4) {
            const int kk = k0 + 2 * half;
            v2f a, b0, b1;
            const float2 pa = *(const float2*)&Pw[l16 * PSTRIDE + kk];
            a[0]  = pa.x;
            a[1]  = pa.y;
            b0[0] = Ys[(kbase + kk) * YSTRIDE + l16];
            b0[1] = Ys[(kbase + kk + 1) * YSTRIDE + l16];
            b1[0] = Ys[(kbase + kk) * YSTRIDE + 16 + l16];
            b1[1] = Ys[(kbase + kk + 1) * YSTRIDE + 16 + l16];
            o0 = __builtin_amdgcn_wmma_f32_16x16x4_f32(false, a, false, b0,
                                                       (short)0, o0, false, false);
            o1 = __builtin_amdgcn_wmma_f32_16x16x4_f32(false, a, false, b1,
                                                       (short)0, o1, false, false);
        }
    }

    // ---- epilogue: 3 extra concatenated logits per row + normalize + store
    #pragma unroll
    for (int v = 0; v < 8; ++v) {
        const int row = qbase + v + 8 * half;    // group-local row 0..1023
        float e[3];
        const int d2 = l16 * 2;
        #pragma unroll
        for (int k = 0; k < 3; ++k) {
            const int nr = row + k - 1;
            const bool ok = (nr >= 0) && (nr < MROWS);
            const int nrc = ok ? nr : row;       // safe address
            const float msk = ok ? 1.0f : 0.0f;
            float part = msk * (Ys[row * YSTRIDE + d2]     * Ys[nrc * YSTRIDE + d2]
                              + Ys[row * YSTRIDE + d2 + 1] * Ys[nrc * YSTRIDE + d2 + 1]);
            #pragma unroll
            for (int mk = 1; mk < 16; mk <<= 1)
                part += __shfl_xor(part, mk, 32);
            e[k] = part * scale;                 // exactly 0 for padded patch
        }
        const float newm = fmaxf(fmaxf(mrow[v], e[0]), fmaxf(e[1], e[2]));
        const float corr = __expf(mrow[v] - newm);
        float psum = 0.0f, add0 = 0.0f, add1 = 0.0f;
        #pragma unroll
        for (int k = 0; k < 3; ++k) {
            const float p = __expf(e[k] - newm);
            psum += p;
            const int nr = row + k - 1;
            const bool ok = (nr >= 0) && (nr < MROWS);
            const int nrc = ok ? nr : row;
            const float msk = ok ? 1.0f : 0.0f;
            add0 += (p * msk) * Ys[nrc * YSTRIDE + l16];
            add1 += (p * msk) * Ys[nrc * YSTRIDE + 16 + l16];
        }
        const float inv = 1.0f / (lrow[v] * corr + psum);
        const float r0 = (o0[v] * corr + add0) * inv;
        const float r1 = (o1[v] * corr + add1) * inv;

        const int n = 2 * row + r;               // sequence position
        if (n < SEQ) {
            const size_t base = ((size_t)(bb * SEQ + n)) * DMODEL + h * DK;
            out[base + l16]      = r0;
            out[base + 16 + l16] = r1;
        }
    }
}

// ---------------------------------------------------------------------------
extern "C" void kernel_launch(void* const* d_in, const int* in_sizes, int n_in,
                              void* d_out, int out_size, void* d_ws, size_t ws_size,
                              hipStream_t stream) {
    const float* x = (const float*)d_in[0];   // (4, 2046, 256)
    const float* W = (const float*)d_in[1];   // (256, 256)
    const float* b = (const float*)d_in[2];   // (256,)
    float* out = (float*)d_out;               // (4, 2046, 256)
    float* qkv = (float*)d_ws;                // (4, 2048, 256) = 8 MB scratch

    dim3 g1(512, 16);                         // 8192/16 x 256/16 tiles
    qkv_gemm_kernel<<<g1, 32, 0, stream>>>(x, W, b, qkv);

    const size_t ldsBytes = (size_t)(MROWS * YSTRIDE + 8 * 16 * PSTRIDE) * sizeof(float);
    dim3 g2(8, 64);                           // 8 query chunks x 64 (b,h,r) groups
    attn_kernel<<<g2, 256, ldsBytes, stream>>>(qkv, out);
}